// SystemIDModel_61718680044067
// MI455X (gfx1250) — compile-verified
//
#include <hip/hip_runtime.h>
#include <math.h>

// ---------------------------------------------------------------------------
// SystemID SSM scan for MI455X (gfx1250, wave32).
//   x1[k+1] = A00*x1[k] + B0*u[k]   (A00 = exp(-dt/(Rp*Cp)), B0 = Rp*(1-A00))
//   x2[k+1] = x2[k]    + B1*u[k]   (B1 = -eta*dt/Qn)
//   y[k] = poly8(x2[k]) - x1[k] - R0*u[k]   (states BEFORE update)
//
// Hierarchy: wave-chunk (2048) -> tile (256) -> segment (16).
// Segment scans = 16x16 lower-triangular matmuls via chained
// V_WMMA_F32_16X16X4_F32. Tile staging/writeback via the Tensor Data Mover
// (TENSOR_LOAD_TO_LDS / TENSOR_STORE_FROM_LDS, TENSORcnt-synchronized).
// ---------------------------------------------------------------------------

typedef __attribute__((ext_vector_type(2))) float v2f;
typedef __attribute__((ext_vector_type(8))) float v8f;
typedef __attribute__((ext_vector_type(4))) unsigned int v4u;
typedef __attribute__((ext_vector_type(8))) int v8i_;
typedef __attribute__((ext_vector_type(4))) int v4i_;

#define DELTA_T     1.0f
#define ETA         1.0f
#define QN          (1.5f * 3600.0f)
#define INITIAL_X2  1.0f

#define CHUNKW      2048          // elements per wave
#define TILES_PER_W 8             // 8 tiles of 256 per wave
#define WAVES_PER_B 8             // 256-thread blocks

#if __has_builtin(__builtin_amdgcn_tensor_load_to_lds) && \
    __has_builtin(__builtin_amdgcn_tensor_store_from_lds)
#define USE_TDM 1
#else
#define USE_TDM 0
#endif

// ---------------------------------------------------------------------------
// Kernel A: per-wave aggregates.
//   wsW[g] = sum_j A00^(C-1-j) * u[g*C + j]   (C = CHUNKW, all weights <= 1)
//   wsS[g] = sum_j u[g*C + j]
// ---------------------------------------------------------------------------
__global__ void sysid_aggr_kernel(const float* __restrict__ uk,
                                  const float* __restrict__ Rp,
                                  const float* __restrict__ Cp,
                                  float* __restrict__ wsW,
                                  float* __restrict__ wsS) {
  const int lane = threadIdx.x & 31;
  const int g = blockIdx.x * WAVES_PER_B + (threadIdx.x >> 5);
  const long base = (long)g * CHUNKW;

  const float dti  = DELTA_T / (Rp[0] * Cp[0]);
  const float A00  = expf(-dti);
  const float A2   = A00 * A00;
  const float A3   = A2 * A00;
  const float A128 = expf(-128.0f * dti);

  const float4* gp = (const float4*)(uk + base);
  float wacc = 0.0f, sacc = 0.0f;
#pragma unroll 4
  for (int m = 0; m < 16; ++m) {
    float4 u4 = gp[m * 32 + lane];                    // elem j = 128m + 4*lane + c
    float pk = A3 * u4.x + A2 * u4.y + A00 * u4.z + u4.w;
    wacc = A128 * wacc + pk;                          // ratio per 128-elem step
    sacc += u4.x + u4.y + u4.z + u4.w;
  }
  float wv = expf(-4.0f * (float)(31 - lane) * dti) * wacc;

#pragma unroll
  for (int o = 16; o >= 1; o >>= 1) {
    wv   += __shfl_xor(wv, o, 32);
    sacc += __shfl_xor(sacc, o, 32);
  }
  if (lane == 0) { wsW[g] = wv; wsS[g] = sacc; }
}

// ---------------------------------------------------------------------------
// Kernel B: one wave scans NW wave-aggregates (exclusive carries).
// ---------------------------------------------------------------------------
__global__ void sysid_scan_kernel(const float* __restrict__ Rp,
                                  const float* __restrict__ Cp,
                                  const float* __restrict__ wsW,
                                  const float* __restrict__ wsS,
                                  float* __restrict__ wsC1,
                                  float* __restrict__ wsC2,
                                  int NW) {
  const int lane = threadIdx.x & 31;
  const float dti   = DELTA_T / (Rp[0] * Cp[0]);
  const float A00   = expf(-dti);
  const float B0    = Rp[0] * (1.0f - A00);
  const float A2048 = expf(-(float)CHUNKW * dti);
  const int   L     = NW / 32;
  const int   base  = lane * L;

  float W = 0.0f, S = 0.0f;
  for (int m = 0; m < L; ++m) {
    int idx = base + m;
    if (idx < NW) { W = A2048 * W + wsW[idx]; S += wsS[idx]; }
  }
  float s1 = B0 * W, s2 = S;

#pragma unroll
  for (int oi = 0; oi < 5; ++oi) {
    int o = 1 << oi;
    float rw = expf(-(float)CHUNKW * (float)L * (float)o * dti);
    float g1 = __shfl_up(s1, o, 32);
    float g2 = __shfl_up(s2, o, 32);
    if (lane >= o) { s1 += rw * g1; s2 += g2; }
  }
  float c1 = __shfl_up(s1, 1, 32); c1 = (lane >= 1) ? c1 : 0.0f;
  float c2 = __shfl_up(s2, 1, 32); c2 = (lane >= 1) ? c2 : 0.0f;

  float x1 = c1, x2 = c2;
  for (int m = 0; m < L; ++m) {
    int idx = base + m;
    if (idx < NW) {
      wsC1[idx] = x1; wsC2[idx] = x2;
      x1 = A2048 * x1 + B0 * wsW[idx];
      x2 += wsS[idx];
    }
  }
}

// ---------------------------------------------------------------------------
// TDM descriptor helpers (ISA cdna5 §8): 1-D tile of 256 f32.
// ---------------------------------------------------------------------------
#if USE_TDM
__device__ __forceinline__ v4u tdm_group0(unsigned lds_addr, unsigned long long ga) {
  v4u g0;
  g0.x = 1u;                                          // count=1, no gather
  g0.y = lds_addr;                                    // lds_addr [63:32]
  g0.z = (unsigned)(ga & 0xffffffffull);              // global_addr [95:64]
  g0.w = (unsigned)((ga >> 32) & 0x01ffffffull) | (2u << 30);  // [120:96], type=2
  return g0;
}

__device__ __forceinline__ v8i_ tdm_group1(void) {
  v8i_ g1;
  g1[0] = (2 << 16);      // workgroup_mask=0, data_size=2 (4B), no pad/iterate
  g1[1] = (256 << 16);    // tensor_dim0 = 256 (bits 79:48 -> dw1[31:16])
  g1[2] = (1 << 16);      // tensor_dim1 = 1   (bits 111:80 -> dw2[31:16])
  g1[3] = (256 << 16);    // tile_dim0 = 256   (bits 127:112)
  g1[4] = 1;              // tile_dim1 = 1
  g1[5] = 256;            // tensor_dim0_stride = 256
  g1[6] = 0;
  g1[7] = 0;
  return g1;
}
#endif

// ---------------------------------------------------------------------------
// Kernel C: replay chunks; TDM staging + WMMA segment scans + epilogue.
// ---------------------------------------------------------------------------
__global__ void sysid_apply_kernel(const float* __restrict__ uk,
                                   const float* __restrict__ Rp,
                                   const float* __restrict__ Cp,
                                   const float* __restrict__ R0p,
                                   const float* __restrict__ ac,
                                   const float* __restrict__ wsC1,
                                   const float* __restrict__ wsC2,
                                   float* __restrict__ out) {
  __shared__ float sdata[WAVES_PER_B][256];

  const int lane = threadIdx.x & 31;
  const int wid  = threadIdx.x >> 5;
  const int g    = blockIdx.x * WAVES_PER_B + wid;
  const long gbase = (long)g * CHUNKW;

  const int m = lane & 15;          // row (A) / column n (B,C,D)
  const int h = lane >> 4;          // lane half

  const float RpCp = Rp[0] * Cp[0];
  const float dti  = DELTA_T / RpCp;
  const float A00  = expf(-dti);
  const float B0   = Rp[0] * (1.0f - A00);
  const float B1   = -ETA * DELTA_T / QN;
  const float R0v  = R0p[0];

  float a_[9];
#pragma unroll
  for (int i = 0; i < 9; ++i) a_[i] = ac[i];

  float pwv[8];
  pwv[0] = 1.0f;
#pragma unroll
  for (int v = 1; v < 8; ++v) pwv[v] = pwv[v - 1] * A00;
  const float A8   = pwv[7] * A00;
  const float A16  = A8 * A8;
  float A16o[4]; A16o[0] = A16;
#pragma unroll
  for (int i = 1; i < 4; ++i) A16o[i] = A16o[i - 1] * A16o[i - 1]; // A16^{1,2,4,8}
  const float A256 = A16o[3] * A16o[3];
  const float a16n = expf(-16.0f * (float)m * dti);   // A00^(16*n)

  // A operands (16x4 slices of the 16x16 scan matrices), ISA layout:
  // lane row = m; VGPR0 holds K = c0 = 4kk+2h, VGPR1 holds K = c0+1.
  v2f a1[4], a2[4];
#pragma unroll
  for (int kk = 0; kk < 4; ++kk) {
    int c0 = 4 * kk + 2 * h;
    int c1c = c0 + 1;
    a1[kk].x = (c0  < m) ? B0 * expf(-(float)(m - 1 - c0 ) * dti) : 0.0f;
    a1[kk].y = (c1c < m) ? B0 * expf(-(float)(m - 1 - c1c) * dti) : 0.0f;
    a2[kk].x = (c0  < m) ? B1 : 0.0f;
    a2[kk].y = (c1c < m) ? B1 : 0.0f;
  }

  float X1w = wsC1[g];
  float X2w = INITIAL_X2 + B1 * wsC2[g];

  float* sw = &sdata[wid][0];
#if USE_TDM
  const unsigned swoff = (unsigned)(uintptr_t)(void*)sw;   // wave-relative LDS byte addr
  const v8i_ g1d = tdm_group1();
  const v4i_ gz = {0, 0, 0, 0};
  const v8i_ gz8 = {0, 0, 0, 0, 0, 0, 0, 0};
#else
  float4* sp4 = (float4*)sw;
#endif

  for (int t = 0; t < TILES_PER_W; ++t) {
    const long tbase = gbase + (long)t * 256;

    // ---- Stage tile into LDS ------------------------------------------------
#if USE_TDM
    {
      v4u g0 = tdm_group0(swoff, (unsigned long long)(uintptr_t)(uk + tbase));
      __builtin_amdgcn_tensor_load_to_lds(g0, g1d, gz, gz, gz8, 0);
    }
    if (t + 1 < TILES_PER_W)
      __builtin_prefetch(uk + tbase + 256 + lane * 8, 0, 3);
    __builtin_amdgcn_s_wait_tensorcnt(0);
    asm volatile("" ::: "memory");       // keep ds reads below the TDM wait
#else
    {
      const float4* gp = (const float4*)(uk + tbase);
      float4 ua = gp[lane * 2];
      float4 ub = gp[lane * 2 + 1];
      sp4[lane * 2]     = ua;
      sp4[lane * 2 + 1] = ub;
    }
    if (t + 1 < TILES_PER_W)
      __builtin_prefetch(uk + tbase + 256 + lane * 8, 0, 3);
    __syncthreads();
#endif

    // ---- B operands from LDS: b[kk] = U[4kk+2h .. +1][n], n = m -------------
    v2f b[4];
#pragma unroll
    for (int kk = 0; kk < 4; ++kk) {
      int idx = m * 16 + 4 * kk + 2 * h;
      b[kk].x = sw[idx];
      b[kk].y = sw[idx + 1];
    }

    // ---- Exclusive segment scans: 4 chained K=4 WMMAs per matrix ------------
    v8f acc1 = {0.f,0.f,0.f,0.f,0.f,0.f,0.f,0.f};
    v8f acc2 = {0.f,0.f,0.f,0.f,0.f,0.f,0.f,0.f};
#pragma unroll
    for (int kk = 0; kk < 4; ++kk) {
      acc1 = __builtin_amdgcn_wmma_f32_16x16x4_f32(
          false, a1[kk], false, b[kk], (short)0, acc1, false, false);
      acc2 = __builtin_amdgcn_wmma_f32_16x16x4_f32(
          false, a2[kk], false, b[kk], (short)0, acc2, false, false);
    }

    // ---- Segment totals (lanes 16..31 hold segment n = lane-16) -------------
    float u15 = sw[m * 16 + 15];
    float t1  = A00 * acc1[7] + B0 * u15;   // x1 increment of full segment
    float t2  = acc2[7] + B1 * u15;         // B1-scaled segment sum

    const int sl = (int)lane - 16;
    float s1 = t1, s2 = t2;
#pragma unroll
    for (int oi = 0; oi < 4; ++oi) {
      int o = 1 << oi;
      float g1v = __shfl_up(s1, o, 32);
      float g2v = __shfl_up(s2, o, 32);
      if (sl >= o) { s1 += A16o[oi] * g1v; s2 += g2v; }
    }
    float c1v = __shfl_up(s1, 1, 32); c1v = (sl >= 1) ? c1v : 0.0f;
    float c2v = __shfl_up(s2, 1, 32); c2v = (sl >= 1) ? c2v : 0.0f;

    float C1  = __shfl(c1v, 16 + m, 32);
    float C2  = __shfl(c2v, 16 + m, 32);
    float Tt1 = __shfl(s1, 31, 32);
    float Tt2 = __shfl(s2, 31, 32);

    float x1seg = a16n * X1w + C1;          // x1 at start of segment n
    float x2seg = X2w + C2;                 // x2 at start of segment n

    // ---- Epilogue: rows i = v + 8h, column n = m; element = 16n + i ---------
#pragma unroll
    for (int v = 0; v < 8; ++v) {
      float rowfac = pwv[v] * (h ? A8 : 1.0f);      // A00^i
      float x1 = acc1[v] + rowfac * x1seg;
      float x2 = x2seg + acc2[v];
      int eidx = m * 16 + v + 8 * h;
      float u  = sw[eidx];
      float p = a_[8];
#pragma unroll
      for (int q = 7; q >= 0; --q) p = p * x2 + a_[q];
      sw[eidx] = p - x1 - R0v * u;                  // y back into LDS
    }

    // ---- Write tile back ----------------------------------------------------
#if USE_TDM
    asm volatile("s_wait_dscnt 0x0" ::: "memory");  // y visible to TDM engine
    {
      v4u g0 = tdm_group0(swoff, (unsigned long long)(uintptr_t)(out + tbase));
      __builtin_amdgcn_tensor_store_from_lds(g0, g1d, gz, gz, gz8, 0);
    }
    __builtin_amdgcn_s_wait_tensorcnt(0);           // LDS reusable next tile
#else
    __syncthreads();
    {
      float4 o0 = sp4[lane * 2];
      float4 o1 = sp4[lane * 2 + 1];
      float4* op = (float4*)(out + tbase);
      op[lane * 2]     = o0;
      op[lane * 2 + 1] = o1;
    }
#endif

    X1w = A256 * X1w + Tt1;
    X2w = X2w + Tt2;
  }
}

// ---------------------------------------------------------------------------
extern "C" void kernel_launch(void* const* d_in, const int* in_sizes, int n_in,
                              void* d_out, int out_size, void* d_ws, size_t ws_size,
                              hipStream_t stream) {
  const float* uk = (const float*)d_in[0];
  const float* Rp = (const float*)d_in[1];
  const float* Cp = (const float*)d_in[2];
  const float* R0 = (const float*)d_in[3];
  const float* ac = (const float*)d_in[4];
  float* out = (float*)d_out;

  const int N  = in_sizes[0];
  const int NW = N / CHUNKW;              // waves (N = 4194304 -> 2048)
  const int nb = NW / WAVES_PER_B;        // 256-thread blocks

  float* wsf  = (float*)d_ws;
  float* wsW  = wsf;
  float* wsS  = wsf + NW;
  float* wsC1 = wsf + 2 * NW;
  float* wsC2 = wsf + 3 * NW;

  sysid_aggr_kernel<<<nb, 256, 0, stream>>>(uk, Rp, Cp, wsW, wsS);
  sysid_scan_kernel<<<1, 32, 0, stream>>>(Rp, Cp, wsW, wsS, wsC1, wsC2, NW);
  sysid_apply_kernel<<<nb, 256, 0, stream>>>(uk, Rp, Cp, R0, ac, wsC1, wsC2, out);
}